// GNNFeatureExtractor_69896297775675
// MI455X (gfx1250) — compile-verified
//
#include <hip/hip_runtime.h>
#include <hip/hip_bf16.h>

typedef __attribute__((ext_vector_type(2))) float v2f;
typedef __attribute__((ext_vector_type(8))) float v8f;

// ---------------- degree / norm kernels ----------------

__global__ void k_fill1(float* __restrict__ deg, int n) {
    int i = blockIdx.x * blockDim.x + threadIdx.x;
    if (i < n) deg[i] = 1.0f;               // self-loop contributes 1 to every node
}

__global__ void k_degcount(const long long* __restrict__ dst, float* __restrict__ deg, int e) {
    int i = blockIdx.x * blockDim.x + threadIdx.x;
    if (i < e) atomicAdd(&deg[(int)dst[i]], 1.0f);
}

__global__ void k_rsqrt(float* __restrict__ deg, int n) {
    int i = blockIdx.x * blockDim.x + threadIdx.x;
    if (i < n) deg[i] = rsqrtf(deg[i]);     // deg >= 1 always (self-loops)
}

// ---------------- WMMA f32 GEMM: out[N,64] = A[N,K] @ W[K,64] ----------------
// One wave computes one 16x16 tile via V_WMMA_F32_16X16X4_F32.
// A-frag (16x4 f32): lanes 0-15 -> {K=k,k+1}, lanes 16-31 -> {K=k+2,k+3}, M = lane&15.
// B-frag (4x16 f32): mirrored; N = lane&15.
// D (16x16 f32): vgpr j holds row j (lanes 0-15) and row 8+j (lanes 16-31).

template <int K>
__global__ void k_gemm_wmma(const float* __restrict__ A, const float* __restrict__ W,
                            float* __restrict__ out) {
    const int lane = threadIdx.x & 31;
    const int wave = threadIdx.x >> 5;      // 4 waves -> cols 0..63
    const int half = lane >> 4;             // 0 or 1
    const int l15  = lane & 15;
    const int row0 = blockIdx.x * 16;
    const int col0 = wave * 16;

    const float* arow = A + (size_t)(row0 + l15) * K;
    const float* wcol = W + col0 + l15;

    v8f acc = {};
#pragma unroll
    for (int k = 0; k < K; k += 4) {
        v2f a;
        a.x = arow[k + 2 * half];
        a.y = arow[k + 2 * half + 1];
        v2f b;
        b.x = wcol[(size_t)(k + 2 * half) * 64];
        b.y = wcol[(size_t)(k + 2 * half + 1) * 64];
        acc = __builtin_amdgcn_wmma_f32_16x16x4_f32(
            /*neg_a=*/false, a, /*neg_b=*/false, b,
            /*c_mod=*/(short)0, acc, /*reuse_a=*/false, /*reuse_b=*/false);
    }

    float* orow = out + (size_t)(row0 + 8 * half) * 64 + col0 + l15;
#pragma unroll
    for (int j = 0; j < 8; ++j)
        orow[(size_t)j * 64] = acc[j];
}

// ---------------- self-loop init: out[i,f] = h[i,f] * dinv[i]^2 ----------------

__global__ void k_selfinit(const float* __restrict__ h, const float* __restrict__ dinv,
                           float* __restrict__ out, int total) {
    int i = blockIdx.x * blockDim.x + threadIdx.x;
    if (i < total) {
        float dv = dinv[i >> 6];
        out[i] = h[i] * dv * dv;
    }
}

// ---------------- edge scatter: out[dst] += h[src] * dinv[src]*dinv[dst] ----------------
// 16 lanes per edge, float4 per lane (64 floats per row).

__global__ void k_edge_agg(const long long* __restrict__ src, const long long* __restrict__ dst,
                           const float* __restrict__ h, const float* __restrict__ dinv,
                           float* __restrict__ out, int e) {
    int t  = blockIdx.x * blockDim.x + threadIdx.x;
    int ed = t >> 4;
    if (ed >= e) return;
    int c  = (t & 15) * 4;
    int s  = (int)src[ed];
    int d  = (int)dst[ed];
    float norm = dinv[s] * dinv[d];
    const float4 v = *(const float4*)(h + (size_t)s * 64 + c);
    float* o = out + (size_t)d * 64 + c;
    atomicAdd(o + 0, v.x * norm);
    atomicAdd(o + 1, v.y * norm);
    atomicAdd(o + 2, v.z * norm);
    atomicAdd(o + 3, v.w * norm);
}

// ---------------- bias + relu (in place) ----------------

__global__ void k_bias_relu(float* __restrict__ out, const float* __restrict__ b, int total) {
    int i = blockIdx.x * blockDim.x + threadIdx.x;
    if (i < total) {
        float v = out[i] + b[i & 63];
        out[i] = v > 0.0f ? v : 0.0f;
    }
}

// ---------------- driver ----------------

extern "C" void kernel_launch(void* const* d_in, const int* in_sizes, int n_in,
                              void* d_out, int out_size, void* d_ws, size_t ws_size,
                              hipStream_t stream) {
    const float*     x  = (const float*)d_in[0];          // [N,128]
    const long long* ei = (const long long*)d_in[1];      // [2,E] int64
    const float*     W1 = (const float*)d_in[2];          // [128,64]
    const float*     b1 = (const float*)d_in[3];          // [64]
    const float*     W2 = (const float*)d_in[4];          // [64,64]
    const float*     b2 = (const float*)d_in[5];          // [64]
    float*           out = (float*)d_out;                 // [N,64]

    const int N = in_sizes[0] / 128;                      // 100000
    const int E = in_sizes[1] / 2;                        // 1600000
    const int NF = N * 64;

    const long long* src = ei;
    const long long* dst = ei + E;

    char*  ws   = (char*)d_ws;
    float* dinv = (float*)ws;
    size_t off0 = ((size_t)N * 4 + 255) & ~(size_t)255;
    size_t hbuf = ((size_t)NF * 4 + 255) & ~(size_t)255;
    float* bufA = (float*)(ws + off0);                    // h_raw buffer
    float* bufB = (float*)(ws + off0 + hbuf);             // aggregation buffer

    const int T = 256;
    // 1) degrees -> dinv
    k_fill1   <<<(N + T - 1) / T, T, 0, stream>>>(dinv, N);
    k_degcount<<<(E + T - 1) / T, T, 0, stream>>>(dst, dinv, E);
    k_rsqrt   <<<(N + T - 1) / T, T, 0, stream>>>(dinv, N);

    // 2) layer 1: h1raw = x @ W1   (N divisible by 16)
    k_gemm_wmma<128><<<N / 16, 128, 0, stream>>>(x, W1, bufA);
    // 3) self-loop term initializes bufB
    k_selfinit<<<(NF + T - 1) / T, T, 0, stream>>>(bufA, dinv, bufB, NF);
    // 4) edge scatter-add
    k_edge_agg<<<((E * 16) + T - 1) / T, T, 0, stream>>>(src, dst, bufA, dinv, bufB, E);
    // 5) h1 = relu(bufB + b1)
    k_bias_relu<<<(NF + T - 1) / T, T, 0, stream>>>(bufB, b1, NF);

    // 6) layer 2: h2raw = h1 @ W2
    k_gemm_wmma<64><<<N / 16, 128, 0, stream>>>(bufB, W2, bufA);
    // 7) self-loop term initializes d_out
    k_selfinit<<<(NF + T - 1) / T, T, 0, stream>>>(bufA, dinv, out, NF);
    // 8) edge scatter-add into d_out
    k_edge_agg<<<((E * 16) + T - 1) / T, T, 0, stream>>>(src, dst, bufA, dinv, out, E);
    // 9) out = relu(out + b2)
    k_bias_relu<<<(NF + T - 1) / T, T, 0, stream>>>(out, b2, NF);
}